// Model_44547400794240
// MI455X (gfx1250) — compile-verified
//
#include <hip/hip_runtime.h>
#include <hip/hip_bf16.h>

typedef __attribute__((ext_vector_type(2))) float v2f;
typedef __attribute__((ext_vector_type(8))) float v8f;

// ===========================================================================
// Graph build: degree histogram -> dinv -> exclusive scan -> CSR fill
// ===========================================================================
__global__ __launch_bounds__(256) void k_zero_i32(int* __restrict__ p, int n) {
    int i = blockIdx.x * blockDim.x + threadIdx.x;
    if (i < n) p[i] = 0;
}

__global__ __launch_bounds__(256) void k_count(const int* __restrict__ dst,
                                               int* __restrict__ count, int n_edges) {
    int e = blockIdx.x * blockDim.x + threadIdx.x;
    if (e < n_edges) atomicAdd(&count[dst[e]], 1);
}

__global__ __launch_bounds__(256) void k_dinv(const int* __restrict__ count,
                                              float* __restrict__ dinv, int n) {
    int i = blockIdx.x * blockDim.x + threadIdx.x;
    if (i < n) dinv[i] = 1.0f / sqrtf((float)(count[i] + 1));  // +1 = self loop
}

// Per-block exclusive scan of 1024 ints (256 thr x 4), emit block sums.
__global__ __launch_bounds__(256) void k_scan_part(const int* __restrict__ count,
                                                   int* __restrict__ part,
                                                   int* __restrict__ bsum, int n) {
    __shared__ int sh[256];
    const int tid = threadIdx.x;
    const int base = blockIdx.x * 1024;
    int vals[4], s = 0;
#pragma unroll
    for (int j = 0; j < 4; ++j) {
        int idx = base + tid * 4 + j;
        vals[j] = (idx < n) ? count[idx] : 0;
        s += vals[j];
    }
    sh[tid] = s;
    __syncthreads();
    for (int off = 1; off < 256; off <<= 1) {
        int v = (tid >= off) ? sh[tid - off] : 0;
        __syncthreads();
        sh[tid] += v;
        __syncthreads();
    }
    int run = sh[tid] - s;  // exclusive prefix of this thread's chunk
#pragma unroll
    for (int j = 0; j < 4; ++j) {
        int idx = base + tid * 4 + j;
        if (idx < n) part[idx] = run;
        run += vals[j];
    }
    if (tid == 255) bsum[blockIdx.x] = sh[255];
}

// Single-block exclusive scan of block sums (nb <= 256).
__global__ __launch_bounds__(256) void k_scan_bsum(int* __restrict__ bsum, int nb) {
    __shared__ int sh[256];
    const int tid = threadIdx.x;
    int v = (tid < nb) ? bsum[tid] : 0;
    sh[tid] = v;
    __syncthreads();
    for (int off = 1; off < 256; off <<= 1) {
        int t = (tid >= off) ? sh[tid - off] : 0;
        __syncthreads();
        sh[tid] += t;
        __syncthreads();
    }
    if (tid < nb) bsum[tid] = sh[tid] - v;
}

__global__ __launch_bounds__(256) void k_scan_add(const int* __restrict__ part,
                                                  const int* __restrict__ bsum,
                                                  int* __restrict__ rowptr,
                                                  int* __restrict__ cursor,
                                                  int n, int n_edges) {
    int i = blockIdx.x * blockDim.x + threadIdx.x;
    if (i < n) {
        int r = part[i] + bsum[i >> 10];
        rowptr[i] = r;
        cursor[i] = r;
    } else if (i == n) {
        rowptr[n] = n_edges;
    }
}

__global__ __launch_bounds__(256) void k_fill(const int* __restrict__ src,
                                              const int* __restrict__ dst,
                                              int* __restrict__ cursor,
                                              int* __restrict__ csr_src, int n_edges) {
    int e = blockIdx.x * blockDim.x + threadIdx.x;
    if (e < n_edges) {
        int slot = atomicAdd(&cursor[dst[e]], 1);
        csr_src[slot] = src[e];
    }
}

// ===========================================================================
// fp32 WMMA GEMM:  C[M x NCOLS] = op(A[M x K]) * B[K x NCOLS]
// One wave computes a 16 x NCOLS strip with V_WMMA_F32_16X16X4_F32.
// Requires M % 16 == 0, K % 4 == 0 (M=100000, K=128).
// ===========================================================================
template <int NCOLS, bool RELU>
__global__ __launch_bounds__(256) void k_gemm_wmma(const float* __restrict__ A,
                                                   const float* __restrict__ B,
                                                   float* __restrict__ C,
                                                   int M, int K) {
    constexpr int NT = NCOLS / 16;
    const int wave = blockIdx.x * (blockDim.x >> 5) + (threadIdx.x >> 5);
    const int lane = threadIdx.x & 31;
    const int half = lane >> 4;   // 0: K={0,1}   1: K={2,3}
    const int l    = lane & 15;   // M (A rows) / N (B,C cols)
    const int m0   = wave * 16;
    if (m0 >= M) return;          // wave-uniform exit: EXEC stays all-1s for WMMA

    v8f acc[NT] = {};

    const float* __restrict__ arow = A + (long long)(m0 + l) * K;
    for (int k0 = 0; k0 < K; k0 += 4) {
        const int ka = k0 + 2 * half;
        float ax = arow[ka];
        float ay = arow[ka + 1];
        if (RELU) { ax = fmaxf(ax, 0.0f); ay = fmaxf(ay, 0.0f); }
        v2f a; a.x = ax; a.y = ay;

        const float* __restrict__ bp = B + (long long)ka * NCOLS + l;
#pragma unroll
        for (int nt = 0; nt < NT; ++nt) {
            v2f b;
            b.x = bp[nt * 16];
            b.y = bp[NCOLS + nt * 16];
            acc[nt] = __builtin_amdgcn_wmma_f32_16x16x4_f32(
                false, a, false, b, (short)0, acc[nt], false, false);
        }
    }

#pragma unroll
    for (int nt = 0; nt < NT; ++nt) {
#pragma unroll
        for (int r = 0; r < 8; ++r) {
            C[(long long)(m0 + r + 8 * half) * NCOLS + nt * 16 + l] = acc[nt][r];
        }
    }
}

// ===========================================================================
// CSR gather aggregation (wave per node, zero float atomics):
//   out[i,:] = bias + dinv[i] * ( dinv[i]*h[i,:] + sum_{s in N(i)} dinv[s]*h[s,:] )
// Edge indices + weights fetched one-per-lane, broadcast via __shfl (wave32).
// ===========================================================================
template <int C>
__global__ __launch_bounds__(256) void k_gather(const float* __restrict__ h,
                                                const int* __restrict__ rowptr,
                                                const int* __restrict__ csr_src,
                                                const float* __restrict__ dinv,
                                                const float* __restrict__ bias,
                                                float* __restrict__ out, int n_nodes) {
    constexpr int VEC = C / 32;  // floats per lane (4 for C=128, 2 for C=64)
    const int wave = blockIdx.x * (blockDim.x >> 5) + (threadIdx.x >> 5);
    const int lane = threadIdx.x & 31;
    if (wave >= n_nodes) return;
    const int node = wave;

    const float di = dinv[node];
    float acc[VEC];
    {
        const float* __restrict__ hr = h + (long long)node * C + lane * VEC;
#pragma unroll
        for (int v = 0; v < VEC; ++v) acc[v] = di * hr[v];   // self loop (x dinv[i] later)
    }

    const int start = rowptr[node];
    const int end   = rowptr[node + 1];
    for (int base = start; base < end; base += 32) {
        const int myidx = base + lane;
        int s = 0; float w = 0.0f;
        if (myidx < end) { s = csr_src[myidx]; w = dinv[s]; }
        const int cnt = (end - base < 32) ? (end - base) : 32;
        for (int j = 0; j < cnt; ++j) {
            const int   sj = __shfl(s, j, 32);
            const float wj = __shfl(w, j, 32);
            const float* __restrict__ hs = h + (long long)sj * C + lane * VEC;
#pragma unroll
            for (int v = 0; v < VEC; ++v) acc[v] = fmaf(wj, hs[v], acc[v]);
        }
    }

    float* __restrict__ op = out + (long long)node * C + lane * VEC;
#pragma unroll
    for (int v = 0; v < VEC; ++v) op[v] = bias[lane * VEC + v] + di * acc[v];
}

// ===========================================================================
// Launch
// ===========================================================================
extern "C" void kernel_launch(void* const* d_in, const int* in_sizes, int n_in,
                              void* d_out, int out_size, void* d_ws, size_t ws_size,
                              hipStream_t stream) {
    constexpr int IN_CH = 128, HID_CH = 128, OUT_CH = 64;

    const float* x  = (const float*)d_in[0];
    const int*   ei = (const int*)d_in[1];
    const float* W1 = (const float*)d_in[2];
    const float* b1 = (const float*)d_in[3];
    const float* W2 = (const float*)d_in[4];
    const float* b2 = (const float*)d_in[5];
    float* out = (float*)d_out;

    const int n_nodes = in_sizes[0] / IN_CH;   // 100000
    const int n_edges = in_sizes[1] / 2;       // 1600000
    const int* src = ei;
    const int* dst = ei + n_edges;

    // --- workspace layout (byte offsets, 512KB-granular slots) ---
    char* base = (char*)d_ws;
    float* dinv    = (float*)(base + 0);                 // N floats
    int*   count   = (int*)  (base + (1u << 19));        // N ints
    int*   cursor  = (int*)  (base + (2u << 19));        // N ints
    int*   rowptr  = (int*)  (base + (3u << 19));        // N+1 ints
    int*   part    = (int*)  (base + (4u << 19));        // N ints
    int*   bsum    = (int*)  (base + (5u << 19));        // <=256 ints
    int*   csr_src = (int*)  (base + (6u << 19));        // E ints (6.4MB)
    float* h1      = (float*)(base + (16ull << 20));     // N*128 floats (51.2MB), reused as h2
    float* agg1    = (float*)(base + (70ull << 20));     // N*128 floats (51.2MB)
    float* h2      = h1;

    const int TB = 256;
    const int nblk_nodes = (n_nodes + TB - 1) / TB;
    const int nblk_edges = (n_edges + TB - 1) / TB;
    const int nb_scan    = (n_nodes + 1023) / 1024;      // 98 (<=256)

    // ---- CSR build -----------------------------------------------------------
    k_zero_i32<<<nblk_nodes, TB, 0, stream>>>(count, n_nodes);
    k_count<<<nblk_edges, TB, 0, stream>>>(dst, count, n_edges);
    k_dinv<<<nblk_nodes, TB, 0, stream>>>(count, dinv, n_nodes);
    k_scan_part<<<nb_scan, TB, 0, stream>>>(count, part, bsum, n_nodes);
    k_scan_bsum<<<1, TB, 0, stream>>>(bsum, nb_scan);
    k_scan_add<<<(n_nodes + 1 + TB - 1) / TB, TB, 0, stream>>>(part, bsum, rowptr, cursor,
                                                               n_nodes, n_edges);
    k_fill<<<nblk_edges, TB, 0, stream>>>(src, dst, cursor, csr_src, n_edges);

    // ---- layer 1: h1 = x @ W1 (WMMA), agg1 = gather ---------------------------
    {
        int blocks = (n_nodes / 16 + 7) / 8;   // 8 waves / block
        k_gemm_wmma<HID_CH, false><<<blocks, TB, 0, stream>>>(x, W1, h1, n_nodes, IN_CH);
    }
    k_gather<HID_CH><<<(n_nodes + 7) / 8, TB, 0, stream>>>(h1, rowptr, csr_src, dinv, b1,
                                                           agg1, n_nodes);

    // ---- layer 2: h2 = relu(agg1) @ W2 (WMMA), out = gather -------------------
    {
        int blocks = (n_nodes / 16 + 7) / 8;
        k_gemm_wmma<OUT_CH, true><<<blocks, TB, 0, stream>>>(agg1, W2, h2, n_nodes, HID_CH);
    }
    k_gather<OUT_CH><<<(n_nodes + 7) / 8, TB, 0, stream>>>(h2, rowptr, csr_src, dinv, b2,
                                                           out, n_nodes);
}